// DecodeBox_12309376270870
// MI455X (gfx1250) — compile-verified
//
#include <hip/hip_runtime.h>
#include <hip/hip_bf16.h>
#include <cstdint>

// ---------------------------------------------------------------------------
// YOLOv3 DecodeBox for MI455X (gfx1250).
// Bandwidth-bound transpose+elementwise (~495 MB traffic -> ~21 us @ 23.3 TB/s).
// Phase 1: one Tensor-Data-Mover tensor_load_to_lds per block gathers the
//          [85 x 64] strided tile into LDS with hardware row padding (stride 65).
// Phase 2: per-attr transform (sigmoid/exp) + fully coalesced NT stores.
// ---------------------------------------------------------------------------

#if __has_builtin(__builtin_amdgcn_tensor_load_to_lds)
#define HAVE_TDM 1
#else
#define HAVE_TDM 0
#endif

#define NATTR    85
#define TILE     64
#define LSTR     65          // LDS row stride (64 data + 1 pad DWORD)
#define NTHREADS 320         // 10 waves; 85*64 = 5440 = 17*320 exactly
#define TOTROWS  22743       // 3*(19^2 + 38^2 + 76^2)

typedef unsigned int v4u __attribute__((ext_vector_type(4)));
typedef int          v8i __attribute__((ext_vector_type(8)));
typedef int          v4i __attribute__((ext_vector_type(4)));

// anchors normalized by input size 608: (anchor/stride)/W == anchor/608
__constant__ float c_awn[9] = {
    10.f/608.f, 16.f/608.f, 33.f/608.f,
    30.f/608.f, 62.f/608.f, 59.f/608.f,
   116.f/608.f,156.f/608.f,373.f/608.f };
__constant__ float c_ahn[9] = {
    13.f/608.f, 30.f/608.f, 23.f/608.f,
    61.f/608.f, 45.f/608.f,119.f/608.f,
    90.f/608.f,198.f/608.f,326.f/608.f };

__device__ __forceinline__ float sigmoidf_(float x) {
    return 1.0f / (1.0f + __expf(-x));
}

template <int W>
__launch_bounds__(NTHREADS)
__global__ void decode_kernel(const float* __restrict__ in, float* __restrict__ out,
                              int scaleBase, int anchorBase) {
    constexpr int   HW   = W * W;
    constexpr float invW = 1.0f / (float)W;

    __shared__ float lds[NATTR * LSTR];

    const int tile = blockIdx.x;
    const int a    = blockIdx.y;   // anchor 0..2
    const int b    = blockIdx.z;   // batch  0..31
    const int tid  = threadIdx.x;
    const int s0   = tile * TILE;

    const float* inbase = in + (long)((b * 3 + a) * NATTR) * HW;

#if HAVE_TDM
    // ---- Phase 1 (TDM): single DMA gathers [85 x 64] tile, HW-padded rows --
    if (tid == 0) {
        const unsigned lds0 = (unsigned)(size_t)(&lds[0]);          // LDS byte offset
        const unsigned long long ga =
            (unsigned long long)(const void*)(inbase + s0);          // tile start (bytes)

        // D# group 0 (128b): count=1 | lds_addr | global_addr[56:0] | type=2
        v4u g0;
        g0.x = 1u;                                                   // count=1, user mode
        g0.y = lds0;                                                 // lds_addr
        g0.z = (unsigned)(ga & 0xffffffffu);                         // global_addr[31:0]
        g0.w = (unsigned)((ga >> 32) & 0x1ffffffu) | (2u << 30);     // addr[56:32] | type=2

        // D# group 1 (256b)
        const unsigned td0 = (unsigned)(HW - s0);   // extent from tile start (tail -> zeros)
        v8i g1;
        g1[0] = (int)((2u << 16)      // data_size = 4 bytes
                    | (1u << 20)      // pad_enable
                    | (5u << 22));    // pad_interval: 2^(5+1)=64 DWORDs; pad_amount=0 -> 1 DWORD
        g1[1] = (int)((td0 & 0xffffu) << 16);                        // tensor_dim0[15:0]
        g1[2] = (int)(((td0 >> 16) & 0xffffu) | ((unsigned)NATTR << 16)); // dim0 hi | tensor_dim1 lo
        g1[3] = (int)((unsigned)TILE << 16);                         // tensor_dim1 hi=0 | tile_dim0=64
        g1[4] = (int)(unsigned)NATTR;                                // tile_dim1=85 | tile_dim2=0
        g1[5] = (int)(unsigned)HW;                                   // tensor_dim0_stride[31:0] = HW
        g1[6] = 0;                                                   // stride0[47:32] | stride1 lo
        g1[7] = 0;                                                   // stride1 hi

        const v4i gz4 = {0, 0, 0, 0};                // groups 2/3 unused (<=2D tensor)
        const v8i gz8 = {0, 0, 0, 0, 0, 0, 0, 0};    // extra operand (6-arg toolchain form)
        __builtin_amdgcn_tensor_load_to_lds(g0, g1, gz4, gz4, gz8, 0);
        __builtin_amdgcn_s_wait_tensorcnt(0);                        // TENSORcnt -> 0
    }
    __syncthreads();
#else
    // ---- Phase 1 (fallback): per-lane async global->LDS, coalesced reads ---
    #pragma unroll 1
    for (int i = tid; i < NATTR * TILE; i += NTHREADS) {
        const int c  = i >> 6;
        const int sl = i & (TILE - 1);
        int s = s0 + sl;
        s = (s < HW) ? s : (HW - 1);
        const float* gptr = inbase + (long)c * HW + s;
        unsigned loff = (unsigned)(size_t)(&lds[c * LSTR + sl]);
        asm volatile("global_load_async_to_lds_b32 %0, %1, off"
                     :: "v"(loff), "v"(gptr)
                     : "memory");
    }
    asm volatile("s_wait_asynccnt 0" ::: "memory");
    __syncthreads();
#endif

    // ---- Phase 2: transform + fully coalesced non-temporal stores ----------
    const float awn = c_awn[anchorBase + a];
    const float ahn = c_ahn[anchorBase + a];

    float* outp = out + (long)b * TOTROWS * NATTR
                      + ((long)scaleBase + (long)a * HW + s0) * NATTR;

    #pragma unroll 1
    for (int j = tid; j < NATTR * TILE; j += NTHREADS) {
        const int sl = j / NATTR;          // const divide -> magic mul
        const int c  = j - sl * NATTR;
        const int s  = s0 + sl;
        if (s < HW) {
            const float v = lds[c * LSTR + sl];
            float r;
            if (c == 0) {
                r = (sigmoidf_(v) + (float)(s % W)) * invW;   // bx / W
            } else if (c == 1) {
                r = (sigmoidf_(v) + (float)(s / W)) * invW;   // by / H (H==W)
            } else if (c == 2) {
                r = __expf(v) * awn;                          // bw = exp * aw/608
            } else if (c == 3) {
                r = __expf(v) * ahn;                          // bh = exp * ah/608
            } else {
                r = sigmoidf_(v);                             // conf / cls
            }
            __builtin_nontemporal_store(r, &outp[j]);         // write-once stream
        }
    }
}

extern "C" void kernel_launch(void* const* d_in, const int* in_sizes, int n_in,
                              void* d_out, int out_size, void* d_ws, size_t ws_size,
                              hipStream_t stream) {
    const float* f0 = (const float*)d_in[0];   // [32,255,19,19]
    const float* f1 = (const float*)d_in[1];   // [32,255,38,38]
    const float* f2 = (const float*)d_in[2];   // [32,255,76,76]
    float* out = (float*)d_out;                // [32,22743,85]

    {
        constexpr int W = 19, nT = (W * W + TILE - 1) / TILE;   // 6 tiles
        decode_kernel<W><<<dim3(nT, 3, 32), NTHREADS, 0, stream>>>(f0, out, 0, 6);
    }
    {
        constexpr int W = 38, nT = (W * W + TILE - 1) / TILE;   // 23 tiles
        decode_kernel<W><<<dim3(nT, 3, 32), NTHREADS, 0, stream>>>(f1, out, 1083, 3);
    }
    {
        constexpr int W = 76, nT = (W * W + TILE - 1) / TILE;   // 91 tiles
        decode_kernel<W><<<dim3(nT, 3, 32), NTHREADS, 0, stream>>>(f2, out, 5415, 0);
    }
}